// RNN_68874095559015
// MI455X (gfx1250) — compile-verified
//
#include <hip/hip_runtime.h>

// ---------------------------------------------------------------------------
// GRU stack: L=2, B=512, T=4096, H=3  (fp32)
// Kernel 1: layer-0 input projection via V_WMMA_F32_16X16X4_F32
// Kernel 2: register-resident fused 2-layer recurrence, 1 lane per sequence
// ---------------------------------------------------------------------------

typedef __attribute__((ext_vector_type(2))) float v2f;
typedef __attribute__((ext_vector_type(8))) float v8f;

#define LNUM 2
#define BSZ  512
#define TSZ  4096
#define HSZ  3
#define GNUM 9      // 3*H
#define GPAD 12     // padded gi0 row stride (floats) -> 48B, 16B aligned

// ---------------------------------------------------------------------------
// Kernel 1: gi0[row, g] = sum_k x[row,k] * Wih0[g,k] + bih0[g]
// rows = b*T + t (natural x order). One 16-row tile per wave, 8 waves/block.
// A (16x4 f32): lanes 0-15 hold {K0,K1}, lanes 16-31 hold {K2,K3=0}
// B (4x16 f32): lanes 0-15 = N, v0/v1 = {K0,K1} (lo half) {K2,K3=0} (hi half)
// C/D (16x16):  lane n = N, vgpr r -> M = r + 8*(lane>=16)
// ---------------------------------------------------------------------------
__global__ __launch_bounds__(256) void gi0_wmma_kernel(
    const float* __restrict__ x,
    const float* __restrict__ Wih0,
    const float* __restrict__ bih0,
    float* __restrict__ gi0)
{
    const int lane = threadIdx.x & 31;
    const int wave = threadIdx.x >> 5;
    const long tile = (long)blockIdx.x * 8 + wave;
    const long row0 = tile * 16;
    const int  m  = lane & 15;
    const int  hi = lane >> 4;

    // A operand: two K-values of x row (row0+m)
    const float* xr = x + (row0 + m) * 3;
    v2f a;
    a.x = (hi == 0) ? xr[0] : xr[2];
    a.y = (hi == 0) ? xr[1] : 0.0f;

    // B operand: Wih0^T, zero-padded to 4x16
    v2f bb;
    bb.x = 0.0f; bb.y = 0.0f;
    if (m < GNUM) {
        bb.x = (hi == 0) ? Wih0[m * 3 + 0] : Wih0[m * 3 + 2];
        bb.y = (hi == 0) ? Wih0[m * 3 + 1] : 0.0f;
    }

    // C operand: bias broadcast along M
    float bias = (m < GNUM) ? bih0[m] : 0.0f;
    v8f c;
#pragma unroll
    for (int r = 0; r < 8; ++r) c[r] = bias;

    v8f d = __builtin_amdgcn_wmma_f32_16x16x4_f32(
        /*neg_a=*/false, a, /*neg_b=*/false, bb,
        /*c_mod=*/(short)0, c, /*reuse_a=*/false, /*reuse_b=*/false);

    if (m < GNUM) {
#pragma unroll
        for (int r = 0; r < 8; ++r) {
            long rowm = row0 + r + 8 * hi;
            gi0[rowm * GPAD + m] = d[r];
        }
    }
}

// ---------------------------------------------------------------------------
// Kernel 2: fused 2-layer recurrence
// ---------------------------------------------------------------------------
__device__ __forceinline__ float fsigm(float v) {
    return __builtin_amdgcn_rcpf(1.0f + __expf(-v));
}
__device__ __forceinline__ float ftanh(float v) {
    float e = __expf(-2.0f * v);
    return (1.0f - e) * __builtin_amdgcn_rcpf(1.0f + e);
}

__device__ __forceinline__ void load_gi(const float* __restrict__ p, float gi[GNUM]) {
    float4 a = *(const float4*)(p);
    float4 b = *(const float4*)(p + 4);
    float  c = p[8];
    gi[0] = a.x; gi[1] = a.y; gi[2] = a.z; gi[3] = a.w;
    gi[4] = b.x; gi[5] = b.y; gi[6] = b.z; gi[7] = b.w;
    gi[8] = c;
}

__device__ __forceinline__ void gru_step(float h[HSZ], const float gi[GNUM],
                                         const float whh[27], const float bhh[GNUM]) {
    float gh[GNUM];
#pragma unroll
    for (int g = 0; g < GNUM; ++g)
        gh[g] = bhh[g] + whh[g*3+0]*h[0] + whh[g*3+1]*h[1] + whh[g*3+2]*h[2];
#pragma unroll
    for (int j = 0; j < HSZ; ++j) {
        float r = fsigm(gi[j]     + gh[j]);
        float z = fsigm(gi[3 + j] + gh[3 + j]);
        float n = ftanh(gi[6 + j] + r * gh[6 + j]);
        h[j] = (1.0f - z) * n + z * h[j];
    }
}

__device__ __forceinline__ float layer1_step(float h1[HSZ], const float o0[HSZ],
                                             const float wih1[27], const float bih1[GNUM],
                                             const float whh1[27], const float bhh1[GNUM],
                                             float wd0, float wd1, float wd2, float bd) {
    float gi[GNUM];
#pragma unroll
    for (int g = 0; g < GNUM; ++g)
        gi[g] = bih1[g] + wih1[g*3+0]*o0[0] + wih1[g*3+1]*o0[1] + wih1[g*3+2]*o0[2];
    gru_step(h1, gi, whh1, bhh1);
    return bd + wd0 * h1[0] + wd1 * h1[1] + wd2 * h1[2];
}

__global__ __launch_bounds__(32) void gru_recurrent_kernel(
    const float* __restrict__ h_init,
    const float* __restrict__ W_ih, const float* __restrict__ W_hh,
    const float* __restrict__ b_ih, const float* __restrict__ b_hh,
    const float* __restrict__ W_dec, const float* __restrict__ b_dec,
    const float* __restrict__ gi0,
    float* __restrict__ out)
{
    const int b = blockIdx.x * 32 + threadIdx.x;

    // Register-resident weights (uniform across lanes)
    float whh0[27], whh1[27], wih1[27];
    float bhh0[GNUM], bhh1[GNUM], bih1[GNUM];
#pragma unroll
    for (int i = 0; i < 27; ++i) {
        whh0[i] = W_hh[i];
        whh1[i] = W_hh[27 + i];
        wih1[i] = W_ih[27 + i];
    }
#pragma unroll
    for (int i = 0; i < GNUM; ++i) {
        bhh0[i] = b_hh[i];
        bhh1[i] = b_hh[GNUM + i];
        bih1[i] = b_ih[GNUM + i];
    }
    const float wd0 = W_dec[0], wd1 = W_dec[1], wd2 = W_dec[2], bd = b_dec[0];

    float h0[HSZ], h1[HSZ], o0[HSZ];
#pragma unroll
    for (int j = 0; j < HSZ; ++j) {
        h0[j] = h_init[b * HSZ + j];
        h1[j] = h_init[BSZ * HSZ + b * HSZ + j];
    }

    const float* girow = gi0 + (long)b * TSZ * GPAD;
    float* yout = out + (long)b * TSZ;

    // t = 0: layer 0 only (layer 1 runs one step behind for ILP)
    {
        float gi[GNUM];
        load_gi(girow, gi);
        gru_step(h0, gi, whh0, bhh0);
        o0[0] = h0[0]; o0[1] = h0[1]; o0[2] = h0[2];
    }

    for (int t = 1; t < TSZ; ++t) {
        float gi[GNUM];
        load_gi(girow + (long)t * GPAD, gi);            // in flight during layer-1 math
        float y = layer1_step(h1, o0, wih1, bih1, whh1, bhh1, wd0, wd1, wd2, bd);
        yout[t - 1] = y;
        gru_step(h0, gi, whh0, bhh0);                   // independent of layer-1 chain
        o0[0] = h0[0]; o0[1] = h0[1]; o0[2] = h0[2];
    }

    // drain the pipeline: layer 1 step for t = T-1
    {
        float y = layer1_step(h1, o0, wih1, bih1, whh1, bhh1, wd0, wd1, wd2, bd);
        yout[TSZ - 1] = y;
    }

    // h_out [L,B,H] appended after y [B,T,1]
#pragma unroll
    for (int j = 0; j < HSZ; ++j) {
        out[(long)BSZ * TSZ + b * HSZ + j] = h0[j];
        out[(long)BSZ * TSZ + BSZ * HSZ + b * HSZ + j] = h1[j];
    }
}

// ---------------------------------------------------------------------------
extern "C" void kernel_launch(void* const* d_in, const int* in_sizes, int n_in,
                              void* d_out, int out_size, void* d_ws, size_t ws_size,
                              hipStream_t stream)
{
    const float* x     = (const float*)d_in[0];
    const float* h     = (const float*)d_in[1];
    const float* W_ih  = (const float*)d_in[2];
    const float* W_hh  = (const float*)d_in[3];
    const float* b_ih  = (const float*)d_in[4];
    const float* b_hh  = (const float*)d_in[5];
    const float* W_dec = (const float*)d_in[6];
    const float* b_dec = (const float*)d_in[7];
    float* out = (float*)d_out;
    float* gi0 = (float*)d_ws;   // B*T*GPAD floats = 100,663,296 bytes

    // Kernel 1: WMMA input projection for layer 0 (B*T/16 tiles, 8 tiles/block)
    const int tiles  = (BSZ * TSZ) / 16;     // 131072
    const int blocks = tiles / 8;            // 16384
    gi0_wmma_kernel<<<blocks, 256, 0, stream>>>(x, W_ih, b_ih, gi0);

    // Kernel 2: one wave per block -> spread 16 waves across WGPs
    gru_recurrent_kernel<<<BSZ / 32, 32, 0, stream>>>(
        h, W_ih, W_hh, b_ih, b_hh, W_dec, b_dec, gi0, out);
}